// PHMSkipConnectAdd_43911745634611
// MI455X (gfx1250) — compile-verified
//
#include <hip/hip_runtime.h>
#include <hip/hip_bf16.h>
#include <math.h>

// MI455X / gfx1250, wave32. fp32 end-to-end; matrix ops via V_WMMA_F32_16X16X4_F32.
// GEMM blocks stage the packed-B column slice in LDS (one column-quad per block,
// 8 row-tiles = 8 waves share it) -> 8x less L2 traffic for B, ds_load in inner loop.

typedef float v2f __attribute__((ext_vector_type(2)));
typedef float v8f __attribute__((ext_vector_type(8)));

#define PQ 4     // phm components
#define DD 49    // per-component dim
#define DF 196   // PQ*DD

static inline int cdiv_ll(long long a, long long b) { return (int)((a + b - 1) / b); }

// ---------------------------------------------------------------------------
// small utility kernels
// ---------------------------------------------------------------------------
__global__ void zero_f32(float* __restrict__ p, int n) {
    int i = blockIdx.x * blockDim.x + threadIdx.x;
    if (i < n) p[i] = 0.f;
}

__global__ void copy_f32(float* __restrict__ dst, const float* __restrict__ src, int n) {
    int i = blockIdx.x * blockDim.x + threadIdx.x;
    if (i < n) dst[i] = src[i];
}

// ---------------------------------------------------------------------------
// atom encoder: h0[n, col] = sum_f atom_emb[f, x[n,f], col]; also h = h0
// ---------------------------------------------------------------------------
__global__ void atom_encoder_kernel(const int* __restrict__ x,
                                    const float* __restrict__ atom_emb,
                                    float* __restrict__ h0, float* __restrict__ h, int N) {
    int gid = blockIdx.x * blockDim.x + threadIdx.x;
    if (gid >= N * DF) return;
    int n = gid / DF, col = gid % DF;
    float v = 0.f;
#pragma unroll
    for (int f = 0; f < 9; ++f) {
        int val = x[n * 9 + f];
        v += atom_emb[((size_t)(f * 119 + val)) * DF + col];
    }
    h0[gid] = v;
    h[gid] = v;
}

// ---------------------------------------------------------------------------
// edge messages: agg[dst] += h[src] + bond_emb(edge_attr)   (agg pre-seeded = h)
// ---------------------------------------------------------------------------
__global__ void edge_message_kernel(const float* __restrict__ h,
                                    const int* __restrict__ edge_index,
                                    const int* __restrict__ edge_attr,
                                    const float* __restrict__ bond_l, // layer slice [3,8,196]
                                    float* __restrict__ agg, int E) {
    int gid = blockIdx.x * blockDim.x + threadIdx.x;
    if (gid >= E * DF) return;
    int e = gid / DF, col = gid % DF;
    int src = edge_index[e];
    int dst = edge_index[E + e];
    float v = h[(size_t)src * DF + col];
#pragma unroll
    for (int f = 0; f < 3; ++f) {
        int val = edge_attr[e * 3 + f];
        v += bond_l[(size_t)(f * 8 + val) * DF + col];
    }
    atomicAdd(&agg[(size_t)dst * DF + col], v);   // global_atomic_add_f32
}

// ---------------------------------------------------------------------------
// build kron-sum PHM matrix: M[(k*din+p)][(j*dout+q)] = sum_i A[i,k,j]*W[i,p,q]
// ---------------------------------------------------------------------------
__global__ void build_phm_matrix(const float* __restrict__ Amat, // [P,P,P]
                                 const float* __restrict__ W,    // [P,din,dout]
                                 float* __restrict__ M, int din, int dout) {
    int C = PQ * dout;
    int total = PQ * din * C;
    int gid = blockIdx.x * blockDim.x + threadIdx.x;
    if (gid >= total) return;
    int kp = gid / C, jq = gid % C;
    int k = kp / din, p = kp % din;
    int j = jq / dout, q = jq % dout;
    float s = 0.f;
#pragma unroll
    for (int i = 0; i < PQ; ++i)
        s += Amat[(i * PQ + k) * PQ + j] * W[((size_t)(i * din + p)) * dout + q];
    M[gid] = s;
}

// ---------------------------------------------------------------------------
// pack dense M [K x C] into wmma B-fragment layout:
//   pack[((s*CTpad + ct)*32 + lane)*2 + {0,1}] = M[4s + (lane<16?0:2) + {0,1}][ct*16 + lane&15]
// (zero-padded for cols >= C)
// ---------------------------------------------------------------------------
__global__ void pack_b_kernel(const float* __restrict__ M, float* __restrict__ pack,
                              int K, int C, int CTpad) {
    int S = K >> 2;
    int total = S * CTpad * 32;
    int gid = blockIdx.x * blockDim.x + threadIdx.x;
    if (gid >= total) return;
    int lane = gid & 31;
    int sc = gid >> 5;
    int ct = sc % CTpad, s = sc / CTpad;
    int kk = s * 4 + ((lane < 16) ? 0 : 2);
    int col = ct * 16 + (lane & 15);
    float v0 = (col < C) ? M[(size_t)kk * C + col] : 0.f;
    float v1 = (col < C) ? M[(size_t)(kk + 1) * C + col] : 0.f;
    pack[(size_t)gid * 2 + 0] = v0;
    pack[(size_t)gid * 2 + 1] = v1;
}

// ---------------------------------------------------------------------------
// fp32 WMMA GEMM: Cout[R x Ccols] = A[R x K] @ M (pre-packed as Bpack)
// One block = 8 waves = 8 consecutive row-tiles, all on the SAME column quad
// (64 cols). The block stages the quad's full B slice (S*256 floats) in LDS
// once; each wave then runs K/4 steps of V_WMMA_F32_16X16X4_F32 with one
// global b64 A-load and four ds b64 B-loads per step.
// ---------------------------------------------------------------------------
__global__ void wmma_gemm_kernel(const float* __restrict__ A,
                                 const float* __restrict__ Bpack,
                                 float* __restrict__ Cout,
                                 int R, int K, int Ccols, int CTpad) {
    extern __shared__ float lds[];                      // S * 256 floats
    const int S = K >> 2;
    const int quads = CTpad >> 2;
    const int rowTiles = (R + 15) >> 4;
    const int cq = blockIdx.x % quads;
    const int rtGroup = blockIdx.x / quads;

    // ---- cooperative stage of B slice for this column quad into LDS ----
    {
        const float4* src = (const float4*)Bpack;
        float4* dst = (float4*)lds;
        const int nvec = S * 64;                        // float4s in slice
        for (int v = threadIdx.x; v < nvec; v += blockDim.x) {
            int s = v >> 6, j = v & 63;
            dst[v] = src[(size_t)(s * CTpad + (cq << 2)) * 16 + j];
        }
    }
    __syncthreads();

    const int lane = threadIdx.x & 31;
    const int rt = rtGroup * 8 + (threadIdx.x >> 5);
    if (rt >= rowTiles) return;                         // after barrier: safe

    int row = rt * 16 + (lane & 15);
    int rowc = (row < R) ? row : (R - 1);               // clamp (R mult of 16 here)
    const float* arow = A + (size_t)rowc * K;
    const int khalf = (lane < 16) ? 0 : 2;

    v8f acc[4];
#pragma unroll
    for (int t = 0; t < 4; ++t) acc[t] = (v8f)0.f;

    const float* lb = lds + lane * 2;
    for (int s = 0; s < S; ++s) {
        v2f a = *(const v2f*)(arow + s * 4 + khalf);
        const float* bp = lb + (s << 8);
        v2f b0 = *(const v2f*)(bp);
        v2f b1 = *(const v2f*)(bp + 64);
        v2f b2 = *(const v2f*)(bp + 128);
        v2f b3 = *(const v2f*)(bp + 192);
        acc[0] = __builtin_amdgcn_wmma_f32_16x16x4_f32(false, a, false, b0, (short)0, acc[0], false, false);
        acc[1] = __builtin_amdgcn_wmma_f32_16x16x4_f32(false, a, false, b1, (short)0, acc[1], false, false);
        acc[2] = __builtin_amdgcn_wmma_f32_16x16x4_f32(false, a, false, b2, (short)0, acc[2], false, false);
        acc[3] = __builtin_amdgcn_wmma_f32_16x16x4_f32(false, a, false, b3, (short)0, acc[3], false, false);
    }

    const int cb = cq * 64 + (lane & 15);
#pragma unroll
    for (int g = 0; g < 8; ++g) {
        int r = rt * 16 + g + ((lane < 16) ? 0 : 8);    // C/D layout: VGPR g -> rows g, g+8
        if (r >= R) continue;
        float* crow = Cout + (size_t)r * Ccols;
        if (cb < Ccols)      crow[cb]      = acc[0][g];
        if (cb + 16 < Ccols) crow[cb + 16] = acc[1][g];
        if (cb + 32 < Ccols) crow[cb + 32] = acc[2][g];
        if (cb + 48 < Ccols) crow[cb + 48] = acc[3][g];
    }
}

// ---------------------------------------------------------------------------
// naive-BN stats: stats[c] += sum x, stats[C+c] += sum x^2   (over rows)
// ---------------------------------------------------------------------------
__global__ void bn_stats_kernel(const float* __restrict__ X, float* __restrict__ stats,
                                int R, int C, int rowsPerBlock) {
    __shared__ float s[512];                            // C <= 256
    for (int i = threadIdx.x; i < 2 * C; i += blockDim.x) s[i] = 0.f;
    __syncthreads();
    long long r0 = (long long)blockIdx.x * rowsPerBlock;
    long long r1 = r0 + rowsPerBlock; if (r1 > R) r1 = R;
    for (long long i = r0 * C + threadIdx.x; i < r1 * C; i += blockDim.x) {
        float v = X[i];
        int c = (int)(i % C);
        atomicAdd(&s[c], v);                            // ds_add_f32
        atomicAdd(&s[C + c], v * v);
    }
    __syncthreads();
    for (int i = threadIdx.x; i < 2 * C; i += blockDim.x) atomicAdd(&stats[i], s[i]);
}

// fused BN -> ReLU -> (+skip), in place
__global__ void bn_apply_kernel(float* __restrict__ X, const float* __restrict__ skip,
                                const float* __restrict__ gamma, const float* __restrict__ beta,
                                const float* __restrict__ stats, int R, int C, float invR) {
    long long gid = (long long)blockIdx.x * blockDim.x + threadIdx.x;
    if (gid >= (long long)R * C) return;
    int c = (int)(gid % C);
    float mu = stats[c] * invR;
    float var = stats[C + c] * invR - mu * mu;
    float v = (X[gid] - mu) * rsqrtf(var + 1e-5f) * gamma[c] + beta[c];
    v = fmaxf(v, 0.f);
    if (skip) v += skip[gid];
    X[gid] = v;
}

// ---------------------------------------------------------------------------
// soft-attention pooling: g = sigmoid(sum_j (lin[n,j,c]+pool_b[j,c])*rt_w[j] + rt_b[c])
// pooled[batch[n], col] += h[n,col] * g[col%49]
// ---------------------------------------------------------------------------
__global__ void pool_gate_scatter(const float* __restrict__ h, const float* __restrict__ lin,
                                  const float* __restrict__ pool_b,
                                  const float* __restrict__ rt_w, const float* __restrict__ rt_b,
                                  const int* __restrict__ batch, float* __restrict__ pooled, int N) {
    int gid = blockIdx.x * blockDim.x + threadIdx.x;
    if (gid >= N * DF) return;
    int n = gid / DF, col = gid % DF, c = col % DD;
    float acc = rt_b[c];
#pragma unroll
    for (int j = 0; j < PQ; ++j)
        acc += (lin[(size_t)n * DF + j * DD + c] + pool_b[j * DD + c]) * rt_w[j];
    float g = 1.f / (1.f + expf(-acc));
    atomicAdd(&pooled[(size_t)batch[n] * DF + col], h[gid] * g);
}

// ---------------------------------------------------------------------------
// fold dn_A[2], dn3_W, rt_w into one 128-vector; V[128] = scalar constant
// ---------------------------------------------------------------------------
__global__ void build_final_vec(const float* __restrict__ A2, const float* __restrict__ W3,
                                const float* __restrict__ dn3_b, const float* __restrict__ rt_w,
                                const float* __restrict__ rt_b, float* __restrict__ V) {
    int c = threadIdx.x;
    if (c < 128) {
        int k = c / 32, p = c % 32;
        float s = 0.f;
#pragma unroll
        for (int i = 0; i < PQ; ++i)
#pragma unroll
            for (int j = 0; j < PQ; ++j)
                s += A2[(i * PQ + k) * PQ + j] * W3[i * 32 + p] * rt_w[j];
        V[c] = s;
    }
    if (c == 0) {
        float cst = rt_b[0];
#pragma unroll
        for (int j = 0; j < PQ; ++j) cst += dn3_b[j] * rt_w[j];
        V[128] = cst;
    }
}

__global__ void final_out_kernel(const float* __restrict__ z2, const float* __restrict__ V,
                                 float* __restrict__ out) {
    __shared__ float s[128];
    int b = blockIdx.x, t = threadIdx.x;
    s[t] = z2[(size_t)b * 128 + t] * V[t];
    __syncthreads();
    for (int off = 64; off > 0; off >>= 1) {
        if (t < off) s[t] += s[t + off];
        __syncthreads();
    }
    if (t == 0) out[b] = s[0] + V[128];
}

// ---------------------------------------------------------------------------
extern "C" void kernel_launch(void* const* d_in, const int* in_sizes, int n_in,
                              void* d_out, int out_size, void* d_ws, size_t ws_size,
                              hipStream_t stream) {
    const int* x          = (const int*)d_in[0];
    const int* edge_index = (const int*)d_in[1];
    const int* edge_attr  = (const int*)d_in[2];
    const int* batch      = (const int*)d_in[3];
    const float* atom_emb = (const float*)d_in[4];
    const float* bond_emb = (const float*)d_in[5];
    const float* conv_A   = (const float*)d_in[6];
    const float* conv_W   = (const float*)d_in[7];
    // d_in[8] conv_b: cancelled exactly by BN mean-subtraction
    const float* mp_gamma = (const float*)d_in[9];
    const float* mp_beta  = (const float*)d_in[10];
    const float* pool_A   = (const float*)d_in[11];
    const float* pool_W   = (const float*)d_in[12];
    const float* pool_b   = (const float*)d_in[13];
    const float* pool_rt_w= (const float*)d_in[14];
    const float* pool_rt_b= (const float*)d_in[15];
    const float* dn_A     = (const float*)d_in[16];
    const float* dn1_W    = (const float*)d_in[17];
    // d_in[18] dn1_b: cancelled by BN
    const float* dn1_gamma= (const float*)d_in[19];
    const float* dn1_beta = (const float*)d_in[20];
    const float* dn2_W    = (const float*)d_in[21];
    // d_in[22] dn2_b: cancelled by BN
    const float* dn2_gamma= (const float*)d_in[23];
    const float* dn2_beta = (const float*)d_in[24];
    const float* dn3_W    = (const float*)d_in[25];
    const float* dn3_b    = (const float*)d_in[26];
    const float* rt_w     = (const float*)d_in[27];
    const float* rt_b     = (const float*)d_in[28];

    const int N  = in_sizes[0] / 9;
    const int E  = in_sizes[1] / 2;
    const int Bn = out_size;                  // graphs (4096)
    const int L  = 3;

    // workspace layout (floats)
    float* ws = (float*)d_ws;
    size_t o = 0;
    float* h0   = ws + o; o += (size_t)N * DF;
    float* h    = ws + o; o += (size_t)N * DF;
    float* tmp  = ws + o; o += (size_t)N * DF;
    float* pooled = ws + o; o += (size_t)Bn * DF;
    float* z1   = ws + o; o += (size_t)Bn * 256;
    float* z2   = ws + o; o += (size_t)Bn * 128;
    float* stats = ws + o; o += 512;
    float* Mdense = ws + o; o += 256 * 256;
    const int packConvSz = 49 * 16 * 64;      // S=49, CTpad=16, 32 lanes * 2
    float* packConv0 = ws + o; o += packConvSz;
    float* packConv1 = ws + o; o += packConvSz;
    float* packConv2 = ws + o; o += packConvSz;
    float* packPool  = ws + o; o += packConvSz;
    float* packM1    = ws + o; o += packConvSz;           // K=196 (S=49), C=256 (CTpad=16)
    float* packM2    = ws + o; o += 64 * 8 * 64;          // K=256 (S=64), C=128 (CTpad=8)
    float* Vfin      = ws + o; o += 132;
    float* packConv[3] = { packConv0, packConv1, packConv2 };

    const dim3 blk(256);

    // ---- precompute kron-sum matrices + wmma B packs ----
    for (int l = 0; l < L; ++l) {
        build_phm_matrix<<<cdiv_ll(DF * DF, 256), blk, 0, stream>>>(
            conv_A + l * 64, conv_W + (size_t)l * PQ * DD * DD, Mdense, DD, DD);
        pack_b_kernel<<<cdiv_ll(49 * 16 * 32, 256), blk, 0, stream>>>(Mdense, packConv[l], DF, DF, 16);
    }
    build_phm_matrix<<<cdiv_ll(DF * DF, 256), blk, 0, stream>>>(pool_A, pool_W, Mdense, DD, DD);
    pack_b_kernel<<<cdiv_ll(49 * 16 * 32, 256), blk, 0, stream>>>(Mdense, packPool, DF, DF, 16);
    build_phm_matrix<<<cdiv_ll(DF * 256, 256), blk, 0, stream>>>(dn_A + 0, dn1_W, Mdense, DD, 64);
    pack_b_kernel<<<cdiv_ll(49 * 16 * 32, 256), blk, 0, stream>>>(Mdense, packM1, DF, 256, 16);
    build_phm_matrix<<<cdiv_ll(256 * 128, 256), blk, 0, stream>>>(dn_A + 64, dn2_W, Mdense, 64, 32);
    pack_b_kernel<<<cdiv_ll(64 * 8 * 32, 256), blk, 0, stream>>>(Mdense, packM2, 256, 128, 8);
    build_final_vec<<<1, 128, 0, stream>>>(dn_A + 128, dn3_W, dn3_b, rt_w, rt_b, Vfin);

    // ---- atom encoder ----
    atom_encoder_kernel<<<cdiv_ll((long long)N * DF, 256), blk, 0, stream>>>(x, atom_emb, h0, h, N);

    // gemm launch helper: blocks = ceil(rowTiles/8) * quads, shmem = (K/4)*256*4 B
    auto gemm_blocks = [](int R, int CTpad) {
        long long rtGroups = ((R + 15) / 16 + 7) / 8;
        return (int)(rtGroups * (CTpad / 4));
    };
    auto gemm_shmem = [](int K) { return (size_t)(K / 4) * 256 * sizeof(float); };

    // ---- message-passing layers ----
    for (int l = 0; l < L; ++l) {
        copy_f32<<<cdiv_ll((long long)N * DF, 256), blk, 0, stream>>>(tmp, h, N * DF);
        edge_message_kernel<<<cdiv_ll((long long)E * DF, 256), blk, 0, stream>>>(
            h, edge_index, edge_attr, bond_emb + (size_t)l * 3 * 8 * DF, tmp, E);
        // z = agg @ Mconv  (written over h; h dead after the scatter)
        wmma_gemm_kernel<<<gemm_blocks(N, 16), blk, gemm_shmem(DF), stream>>>(
            tmp, packConv[l], h, N, DF, DF, 16);
        zero_f32<<<2, blk, 0, stream>>>(stats, 512);
        bn_stats_kernel<<<cdiv_ll(N, 256), blk, 0, stream>>>(h, stats, N, DF, 256);
        bn_apply_kernel<<<cdiv_ll((long long)N * DF, 256), blk, 0, stream>>>(
            h, h0, mp_gamma + l * DF, mp_beta + l * DF, stats, N, DF, 1.f / (float)N);
    }

    // ---- soft-attention pooling ----
    wmma_gemm_kernel<<<gemm_blocks(N, 16), blk, gemm_shmem(DF), stream>>>(
        h, packPool, tmp, N, DF, DF, 16);
    zero_f32<<<cdiv_ll((long long)Bn * DF, 256), blk, 0, stream>>>(pooled, Bn * DF);
    pool_gate_scatter<<<cdiv_ll((long long)N * DF, 256), blk, 0, stream>>>(
        h, tmp, pool_b, pool_rt_w, pool_rt_b, batch, pooled, N);

    // ---- downstream PHM MLP ----
    wmma_gemm_kernel<<<gemm_blocks(Bn, 16), blk, gemm_shmem(DF), stream>>>(
        pooled, packM1, z1, Bn, DF, 256, 16);
    zero_f32<<<2, blk, 0, stream>>>(stats, 512);
    bn_stats_kernel<<<cdiv_ll(Bn, 64), blk, 0, stream>>>(z1, stats, Bn, 256, 64);
    bn_apply_kernel<<<cdiv_ll((long long)Bn * 256, 256), blk, 0, stream>>>(
        z1, nullptr, dn1_gamma, dn1_beta, stats, Bn, 256, 1.f / (float)Bn);

    wmma_gemm_kernel<<<gemm_blocks(Bn, 8), blk, gemm_shmem(256), stream>>>(
        z1, packM2, z2, Bn, 256, 128, 8);
    zero_f32<<<2, blk, 0, stream>>>(stats, 512);
    bn_stats_kernel<<<cdiv_ll(Bn, 64), blk, 0, stream>>>(z2, stats, Bn, 128, 64);
    bn_apply_kernel<<<cdiv_ll((long long)Bn * 128, 256), blk, 0, stream>>>(
        z2, nullptr, dn2_gamma, dn2_beta, stats, Bn, 128, 1.f / (float)Bn);

    final_out_kernel<<<Bn, 128, 0, stream>>>(z2, Vfin, (float*)d_out);
}